// SSD_NeRF_46222438039882
// MI455X (gfx1250) — compile-verified
//
#include <hip/hip_runtime.h>

typedef float v2f __attribute__((ext_vector_type(2)));
typedef float v8f __attribute__((ext_vector_type(8)));

#define CHUNK 2048          // lidar points staged per LDS pass (32 KB)
#define BLOCK 256           // 8 waves of 32
#define ROWS_PER_BLOCK 128  // 8 waves * 16 query rows
#define TILES_PER_CHUNK (CHUNK / 16)

// Nearest-neighbor + gather via V_WMMA_F32_16X16X4_F32.
// Distance surrogate per (m,n): A[m]·B[n], A=[-2qx,-2qy,-2qz,1], B=[kx,ky,kz,|k|^2]
// == -2 q·k + |k|^2 (per-row +|q|^2 constant doesn't change the argmin).
__global__ __launch_bounds__(BLOCK) void nn_wmma_gather_kernel(
    const float* __restrict__ pts,       // (M,3)
    const float* __restrict__ lidar,     // (N,3)
    const float* __restrict__ features,  // (N,D)
    float* __restrict__ outp,            // (M,D)
    int M, int N, int D)
{
    // B fragments pre-swizzled into WMMA lane layout:
    // sB[t*32 + c]      = (kx, ky)      -> lane c      (K0,K1)
    // sB[t*32 + c + 16] = (kz, |k|^2)   -> lane c + 16 (K2,K3)
    __shared__ float2 sB[TILES_PER_CHUNK * 32];

    const int tid  = threadIdx.x;
    const int lane = tid & 31;
    const int wave = tid >> 5;
    const int row  = lane & 15;          // A-row / B-column within a 16-wide tile
    const bool hi  = lane >= 16;         // lanes 16..31 carry K=2,3 halves

    const int mBase = blockIdx.x * ROWS_PER_BLOCK + wave * 16;
    const int m     = mBase + row;

    // ---- A fragment (16x4 f32): lane<16 -> (-2qx,-2qy); lane>=16 -> (-2qz, 1)
    const float qx = pts[(size_t)m * 3 + 0];
    const float qy = pts[(size_t)m * 3 + 1];
    const float qz = pts[(size_t)m * 3 + 2];
    v2f a;
    a[0] = hi ? (-2.0f * qz) : (-2.0f * qx);
    a[1] = hi ? 1.0f         : (-2.0f * qy);

    // Running argmin per C-row r (global row r for lanes 0..15, r+8 for lanes 16..31)
    float bestVal[8];
    int   bestIdx[8];
#pragma unroll
    for (int r = 0; r < 8; ++r) { bestVal[r] = 3.402823466e+38f; bestIdx[r] = 0; }

    const int nChunks = N / CHUNK;
    for (int c = 0; c < nChunks; ++c) {
        const int chunkBase = c * CHUNK;

        // Cooperative stage into B-fragment layout (8 points/thread)
        for (int i = tid; i < CHUNK; i += BLOCK) {
            const int n = chunkBase + i;
            const float x = lidar[(size_t)n * 3 + 0];
            const float y = lidar[(size_t)n * 3 + 1];
            const float z = lidar[(size_t)n * 3 + 2];
            const int t  = i >> 4;
            const int cc = i & 15;
            sB[t * 32 + cc]      = make_float2(x, y);
            sB[t * 32 + cc + 16] = make_float2(z, x * x + y * y + z * z);
        }
        // Prefetch next chunk toward near caches (global_prefetch_b8), one line per thread.
        if (c + 1 < nChunks) {
            __builtin_prefetch(lidar + (size_t)(chunkBase + CHUNK) * 3 + (size_t)tid * 24, 0, 3);
        }
        __syncthreads();

        // One V_WMMA_F32_16X16X4_F32 per 16-wide column tile; b is a single ds_load_b64.
#pragma unroll 4
        for (int t = 0; t < TILES_PER_CHUNK; ++t) {
            const float2 bf = sB[t * 32 + lane];
            v2f b;
            b[0] = bf.x;
            b[1] = bf.y;
            v8f cacc = {0.f, 0.f, 0.f, 0.f, 0.f, 0.f, 0.f, 0.f};
            cacc = __builtin_amdgcn_wmma_f32_16x16x4_f32(
                false, a, false, b, (short)0, cacc, false, false);
            const int nIdx = chunkBase + t * 16 + row;  // this lane's global column
#pragma unroll
            for (int r = 0; r < 8; ++r) {
                const bool lt = cacc[r] < bestVal[r];
                bestIdx[r] = lt ? nIdx    : bestIdx[r];
                bestVal[r] = lt ? cacc[r] : bestVal[r];
            }
        }
        __syncthreads();
    }

    // ---- Cross-lane argmin within each 16-lane half (columns striped across lanes)
#pragma unroll
    for (int r = 0; r < 8; ++r) {
        float v = bestVal[r];
        int   i = bestIdx[r];
#pragma unroll
        for (int off = 1; off < 16; off <<= 1) {
            const float ov = __shfl_xor(v, off, 32);
            const int   oi = __shfl_xor(i, off, 32);
            if (ov < v || (ov == v && oi < i)) { v = ov; i = oi; }  // first-occurrence tiebreak
        }
        bestVal[r] = v;
        bestIdx[r] = i;
    }

    // ---- Gather feature rows: row r (0..7) lives in lanes 0..15, row r+8 in lanes 16..31
    const int dVec = D >> 2;  // float4s per feature row (D=128 -> 32: one per lane)
#pragma unroll
    for (int r = 0; r < 16; ++r) {
        const int nn = __shfl(bestIdx[r & 7], (r < 8) ? 0 : 16, 32);
        const float4* __restrict__ src = (const float4*)(features + (size_t)nn * D);
        float4* __restrict__ dst       = (float4*)(outp + (size_t)(mBase + r) * D);
        for (int d = lane; d < dVec; d += 32) dst[d] = src[d];
    }
}

extern "C" void kernel_launch(void* const* d_in, const int* in_sizes, int n_in,
                              void* d_out, int out_size, void* d_ws, size_t ws_size,
                              hipStream_t stream) {
    const float* pts      = (const float*)d_in[0];  // (1, 1024, 32, 3)
    const float* lidar    = (const float*)d_in[1];  // (1, 8192, 3)
    const float* features = (const float*)d_in[2];  // (1, 8192, 128)
    float* outp           = (float*)d_out;          // (1, 1024, 32, 128)

    const int M = in_sizes[0] / 3;   // 32768 query points
    const int N = in_sizes[1] / 3;   // 8192 lidar points
    const int D = in_sizes[2] / N;   // 128 feature dim

    const int grid = M / ROWS_PER_BLOCK;  // 256 blocks
    nn_wmma_gather_kernel<<<grid, BLOCK, 0, stream>>>(pts, lidar, features, outp, M, N, D);
}